// ESN_mlr_mro_5394478924029
// MI455X (gfx1250) — compile-verified
//
#include <hip/hip_runtime.h>
#include <hip/hip_bf16.h>

// ---------------------------------------------------------------------------
// ESN with mixture-of-low-rank readouts, restructured for MI455X (gfx1250):
//   Phase 1 (sequential): 512 step kernels computing the recurrence h_t.
//                         h_prev staged into LDS by the Tensor Data Mover
//                         (tensor_load_to_lds + s_wait_tensorcnt).
//   Phase 2 (batched)   : gates + top-2 softmax, then WMMA GEMM H x B_w with
//                         probability-weighted epilogue -> r_mix (4096 x 512).
//   Phase 3 (batched)   : WMMA GEMM r_mix x A_w^T + A_b -> logits (B,T,V).
// fp32 throughout (reference is fp32) using V_WMMA_F32_16X16X4_F32.
// ---------------------------------------------------------------------------

#define B_  8
#define T_  512
#define V_  32000
#define N_  2048
#define K_  4
#define R_  512
#define CLIP_ 10.0f

typedef float v2f __attribute__((ext_vector_type(2)));
typedef float v8f __attribute__((ext_vector_type(8)));
typedef unsigned int u32x4 __attribute__((ext_vector_type(4)));
typedef int i32x4 __attribute__((ext_vector_type(4)));
typedef int i32x8 __attribute__((ext_vector_type(8)));

__device__ __forceinline__ v8f wmma_k4(v2f a, v2f b, v8f c) {
    // D = A(16x4,f32) * B(4x16,f32) + C(16x16,f32)
    return __builtin_amdgcn_wmma_f32_16x16x4_f32(
        /*neg_a=*/false, a, /*neg_b=*/false, b,
        /*c_mod=*/(short)0, c, /*reuse_a=*/false, /*reuse_b=*/false);
}

// TDM: DMA a contiguous `bytes/4`-dword 1-row tile from global -> LDS offset 0.
// Descriptor per CDNA5 ISA (2D, data_size=4B).
__device__ __forceinline__ void tdm_load_row_to_lds(const void* gsrc,
                                                    unsigned int nelem_f32) {
    const unsigned long long ga = (unsigned long long)(uintptr_t)gsrc;
    u32x4 g0;
    g0[0] = 1u;                                   // count=1, user descriptor
    g0[1] = 0u;                                   // lds_addr = 0 (hs[] base)
    g0[2] = (unsigned int)(ga & 0xffffffffu);     // global_addr[31:0]
    g0[3] = (unsigned int)((ga >> 32) & 0x01ffffffu) | (2u << 30); // addr[56:32] | type=2
    i32x8 g1;
    g1[0] = (int)(2u << 16);                      // data_size = 4 bytes
    g1[1] = (int)((nelem_f32 & 0xffffu) << 16);   // tensor_dim0[15:0]
    g1[2] = (int)((nelem_f32 >> 16) & 0xffffu) | (int)(1u << 16); // dim0 hi | tensor_dim1=1
    g1[3] = (int)((nelem_f32 & 0xffffu) << 16);   // tile_dim0 = nelem
    g1[4] = 1;                                    // tile_dim1 = 1
    g1[5] = (int)nelem_f32;                       // tensor_dim0_stride (unused, 1 row)
    g1[6] = 0;
    g1[7] = 0;
    i32x4 g2 = {0, 0, 0, 0};
    i32x4 g3 = {0, 0, 0, 0};
#if __clang_major__ >= 23
    i32x8 g4 = {0, 0, 0, 0, 0, 0, 0, 0};
    __builtin_amdgcn_tensor_load_to_lds(g0, g1, g2, g3, g4, 0);
#else
    __builtin_amdgcn_tensor_load_to_lds(g0, g1, g2, g3, 0);
#endif
    __builtin_amdgcn_s_wait_tensorcnt(0);         // TENSORcnt is per-wave
}

// -------------------------------------------------------------- phase 0: a
__global__ void k_sigmoid_a(const float* __restrict__ param_a,
                            float* __restrict__ a_vec) {
    int n = blockIdx.x * blockDim.x + threadIdx.x;
    if (n < N_) a_vec[n] = 1.0f / (1.0f + __expf(-param_a[n]));
}

// -------------------------------------------------------------- phase 1: recurrence
// grid 8 x 256 threads: thread owns column n = bid*256+tid for all 8 batch rows.
// h_prev (8x2048 = 64KB) staged into LDS by TDM (wave 0 issues, all waves sync);
// inner loop streams one coalesced row of W_rec_T per m and does 8 broadcast FMAs.
__global__ void k_step(const int*   __restrict__ x,
                       const float* __restrict__ E_in,
                       const float* __restrict__ W,      // (N,N) row-major
                       const float* __restrict__ a_vec,
                       float*       __restrict__ H,      // (T,B,N)
                       int t) {
    __shared__ float hs[B_ * N_];                        // 64 KB, LDS offset 0
    const int n = blockIdx.x * blockDim.x + threadIdx.x; // global column

    if (t > 0) {
        if (threadIdx.x < 32) {                          // one wave drives the TDM
            tdm_load_row_to_lds(H + (size_t)(t - 1) * B_ * N_, B_ * N_);
        }
    }
    __syncthreads();                                     // tile visible to all waves

    float acc[B_];
#pragma unroll
    for (int b = 0; b < B_; ++b) acc[b] = 0.0f;

    if (t > 0) {
#pragma unroll 4
        for (int m = 0; m < N_; ++m) {
            const float w = W[(size_t)m * N_ + n];       // coalesced
#pragma unroll
            for (int b = 0; b < B_; ++b)
                acc[b] = fmaf(hs[b * N_ + m], w, acc[b]); // LDS broadcast
        }
    }

    const float alpha = a_vec[n];
    float* hcur = H + (size_t)t * B_ * N_;
#pragma unroll
    for (int b = 0; b < B_; ++b) {
        const int   tok = x[b * T_ + t];
        const float u   = E_in[(size_t)tok * N_ + n];
        float pre = u + acc[b];
        pre = fminf(fmaxf(pre, -CLIP_), CLIP_);
        const float hp = (t > 0) ? hs[b * N_ + n] : 0.0f;
        hcur[b * N_ + n] = (1.0f - alpha) * hp + alpha * tanhf(pre);
    }
}

// -------------------------------------------------------------- phase 2a: gates + top-2 softmax
// one wave (32 lanes) per row m = t*B+b; fixed-order shuffle reduction.
__global__ void k_gates(const float* __restrict__ H,       // (T*B, N)
                        const float* __restrict__ gate_w,  // (K, N)
                        const float* __restrict__ gate_b,  // (K,)
                        float*       __restrict__ p) {     // (T*B, K)
    const int wave = threadIdx.x >> 5;
    const int lane = threadIdx.x & 31;
    const int row  = blockIdx.x * (blockDim.x >> 5) + wave;
    if (row >= T_ * B_) return;
    const float* hrow = H + (size_t)row * N_;

    float g[K_];
#pragma unroll
    for (int k = 0; k < K_; ++k) {
        const float* wk = gate_w + (size_t)k * N_;
        float s = 0.0f;
        for (int n = lane; n < N_; n += 32) s = fmaf(hrow[n], wk[n], s);
#pragma unroll
        for (int off = 16; off > 0; off >>= 1) s += __shfl_xor(s, off, 32);
        g[k] = s + gate_b[k];
    }

    if (lane == 0) {
        int i0 = 0;
#pragma unroll
        for (int k = 1; k < K_; ++k) if (g[k] > g[i0]) i0 = k;
        int i1 = (i0 == 0) ? 1 : 0;
#pragma unroll
        for (int k = 0; k < K_; ++k) if (k != i0 && g[k] > g[i1]) i1 = k;
        const float e1 = __expf(g[i1] - g[i0]);          // top value normalized out
        const float inv = 1.0f / (1.0f + e1);
        float pv[K_] = {0.0f, 0.0f, 0.0f, 0.0f};
        pv[i0] = inv;
        pv[i1] = e1 * inv;
#pragma unroll
        for (int k = 0; k < K_; ++k) p[(size_t)row * K_ + k] = pv[k];
    }
}

// -------------------------------------------------------------- phase 2b: r_mix (WMMA)
// one wave per (16-row m-tile, 16-col r-tile); 4 expert accumulators over N=2048;
// epilogue mixes with top-2 probabilities.
__global__ void k_rmix(const float* __restrict__ H,     // (T*B, N)
                       const float* __restrict__ B_w,   // (K, R, N)
                       const float* __restrict__ p,     // (T*B, K)
                       float*       __restrict__ r_mix) // (T*B, R)
{
    const int lane = threadIdx.x & 31;
    const int lm = lane & 15, lh = lane >> 4;
    const int r0 = blockIdx.x * 16;
    const int m0 = blockIdx.y * 16;

    v8f c[K_] = {};
    const float* aptr = H + (size_t)(m0 + lm) * N_ + 2 * lh;
    const float* bptr0 = B_w + (size_t)(r0 + lm) * N_ + 2 * lh;

    for (int n0 = 0; n0 < N_; n0 += 4) {
        const v2f a = *(const v2f*)(aptr + n0);
#pragma unroll
        for (int e = 0; e < K_; ++e) {
            const v2f b = *(const v2f*)(bptr0 + (size_t)e * R_ * N_ + n0);
            c[e] = wmma_k4(a, b, c[e]);
        }
    }

#pragma unroll
    for (int j = 0; j < 8; ++j) {
        const int m = m0 + j + 8 * lh;                   // C/D layout row
        const float4 pv = *(const float4*)(p + (size_t)m * K_);
        const float v = pv.x * c[0][j] + pv.y * c[1][j] +
                        pv.z * c[2][j] + pv.w * c[3][j];
        r_mix[(size_t)m * R_ + r0 + lm] = v;
    }
}

// -------------------------------------------------------------- phase 3: logits (WMMA)
// one wave per (16-row m-tile, 64-col v-tile): A fragment reused across 4 B tiles.
__global__ void k_logits(const float* __restrict__ r_mix, // (T*B, R)
                         const float* __restrict__ A_w,   // (V, R)
                         const float* __restrict__ A_b,   // (V,)
                         float*       __restrict__ out)   // (B, T, V)
{
    const int lane = threadIdx.x & 31;
    const int lm = lane & 15, lh = lane >> 4;
    const int n0 = blockIdx.x * 64;
    const int m0 = blockIdx.y * 16;

    v8f c[4] = {};
    const float* aptr = r_mix + (size_t)(m0 + lm) * R_ + 2 * lh;
    const float* bbase = A_w + (size_t)(n0 + lm) * R_ + 2 * lh;

    for (int k0 = 0; k0 < R_; k0 += 4) {
        const v2f a = *(const v2f*)(aptr + k0);
        __builtin_prefetch((const void*)(bbase + k0 + 64), 0, 1); // global_prefetch
#pragma unroll
        for (int s = 0; s < 4; ++s) {
            const v2f b = *(const v2f*)(bbase + (size_t)s * 16 * R_ + k0);
            c[s] = wmma_k4(a, b, c[s]);
        }
    }

#pragma unroll
    for (int j = 0; j < 8; ++j) {
        const int m = m0 + j + 8 * lh;                   // row index = t*B + b
        const int t = m >> 3;
        const int b = m & 7;
        float* orow = out + (size_t)b * T_ * V_ + (size_t)t * V_;
#pragma unroll
        for (int s = 0; s < 4; ++s) {
            const int v = n0 + s * 16 + lm;
            orow[v] = c[s][j] + A_b[v];
        }
    }
}

// ---------------------------------------------------------------------------
extern "C" void kernel_launch(void* const* d_in, const int* in_sizes, int n_in,
                              void* d_out, int out_size, void* d_ws, size_t ws_size,
                              hipStream_t stream) {
    (void)in_sizes; (void)n_in; (void)out_size; (void)ws_size;
    const int*   x       = (const int*)  d_in[0];
    const float* E_in    = (const float*)d_in[1];
    const float* W_rec_T = (const float*)d_in[2];
    const float* param_a = (const float*)d_in[3];
    const float* B_w     = (const float*)d_in[4];
    const float* A_w     = (const float*)d_in[5];
    const float* A_b     = (const float*)d_in[6];
    const float* gate_w  = (const float*)d_in[7];
    const float* gate_b  = (const float*)d_in[8];
    float* out = (float*)d_out;

    // workspace layout (all 256B aligned)
    char* ws = (char*)d_ws;
    float* a_vec = (float*)(ws);                               //  8 KB
    float* p     = (float*)(ws + 8192);                        // 64 KB
    float* H     = (float*)(ws + 8192 + 65536);                // 32 MB
    float* r_mix = (float*)(ws + 8192 + 65536 + (size_t)T_ * B_ * N_ * 4); // 8 MB

    // phase 0: leak rates
    k_sigmoid_a<<<(N_ + 255) / 256, 256, 0, stream>>>(param_a, a_vec);

    // phase 1: sequential recurrence (T step kernels, cheap inside a graph)
    for (int t = 0; t < T_; ++t)
        k_step<<<N_ / 256, 256, 0, stream>>>(x, E_in, W_rec_T, a_vec, H, t);

    // phase 2a: gate logits + top-2 softmax (one wave per (t,b) row)
    k_gates<<<(T_ * B_) / 8, 256, 0, stream>>>(H, gate_w, gate_b, p);

    // phase 2b: probability-mixed low-rank projection, WMMA f32 16x16x4
    k_rmix<<<dim3(R_ / 16, (T_ * B_) / 16), 32, 0, stream>>>(H, B_w, p, r_mix);

    // phase 3: readout GEMM + bias, WMMA f32 16x16x4, writes (B,T,V) directly
    k_logits<<<dim3(V_ / 64, (T_ * B_) / 16), 32, 0, stream>>>(r_mix, A_w, A_b, out);
}